// QuanvolutionFilter_65481071404034
// MI455X (gfx1250) — compile-verified
//
#include <hip/hip_runtime.h>
#include <math.h>

// ---------------------------------------------------------------------------
// Quanvolution (analytic 4-qubit circuit) for MI455X / gfx1250.
//
// Roofline: 65536*784 f32 in + same out = 411 MB @ 23.3 TB/s => ~17.6 us floor.
// Compute is ~2 hardware transcendentals + ~4 VALU per pixel => deeply
// memory-bound. No contraction dimension exists, so WMMA cannot express this;
// the CDNA5 features that pay are:
//   * global_load_async_to_lds_b128 (ASYNCcnt path): dense 16B HBM reads into
//     LDS, from which the 2x2-patch geometry is served with DS_LOAD_2ADDR_B64
//     (compiler merges both patch rows into ONE ds instruction).
//   * Non-temporal hints both ways: 411 MB streamed once > 192 MB L2, and the
//     input is consumed from LDS, never re-read from cache.
//   * Coalesced B128 NT stores (output offset == base + idx*16B exactly).
// ---------------------------------------------------------------------------

#define THREADS        256
#define IMG_PER_BLOCK  8
#define IMG_FLOATS     784                            // 28*28
#define PATCHES        196                            // 14*14
#define GROUP_FLOATS   (IMG_PER_BLOCK * IMG_FLOATS)   // 6272 floats = 24.5 KB
#define GROUP_CHUNKS   (GROUP_FLOATS / 4)             // 1568 x B128
#define GROUP_PATCHES  (IMG_PER_BLOCK * PATCHES)      // 1568 patches

// Native clang vector (NOT HIP_vector_type) so __builtin_nontemporal_store
// accepts it; lowers to one global_store_b128 with NT policy.
typedef float v4f __attribute__((ext_vector_type(4)));

// ---- tiny precompute: 8 trig constants from params (full-precision libm,
//      arbitrary-range angles; runs once with 4 lanes, cost irrelevant) ------
// params layout: params[0][q] = a_q (RZ), params[1][q] = b_q (RX)
// cs[q]   = cos(b_q)
// cs[4+q] = sin(b_q) * sin(a_q)
__global__ void quanv_precompute(const float* __restrict__ params,
                                 float* __restrict__ cs) {
    int q = threadIdx.x;
    if (q < 4) {
        float a = params[q];
        float b = params[4 + q];
        cs[q]     = cosf(b);
        cs[4 + q] = sinf(b) * sinf(a);
    }
}

__global__ __launch_bounds__(THREADS)
void quanv_main(const float* __restrict__ x,
                const float* __restrict__ cs,
                float* __restrict__ out,
                long long total_floats) {
    __shared__ float smem[GROUP_FLOATS];

    const int tid = threadIdx.x;
    const long long grp_base = (long long)blockIdx.x * GROUP_FLOATS;

    // Uniform trig constants (scalar loads, KMcnt path).
    const float C0 = cs[0], C1 = cs[1], C2 = cs[2], C3 = cs[3];
    const float S0 = cs[4], S1 = cs[5], S2 = cs[6], S3 = cs[7];

    // LDS byte address = low 32 bits of the generic pointer (LDS aperture
    // contributes only the high 32 bits on CDNA5).
    const uint32_t lds_base = (uint32_t)(uintptr_t)&smem[0];

    // ---- Stage this block's contiguous pixel range into LDS with CDNA5
    //      async B128 copies (ASYNCcnt-tracked, no VGPR staging). Input is
    //      touched exactly once -> non-temporal. ----
    for (int c = tid; c < GROUP_CHUNKS; c += THREADS) {
        long long gofs = grp_base + (long long)c * 4;
        if (gofs + 4 <= total_floats) {
            uint64_t gaddr = (uint64_t)(uintptr_t)(x + gofs);
            uint32_t laddr = lds_base + (uint32_t)c * 16u;
            asm volatile("global_load_async_to_lds_b128 %0, %1, off th:TH_LOAD_NT"
                         :: "v"(laddr), "v"(gaddr)
                         : "memory");
        }
    }
#if __has_builtin(__builtin_amdgcn_s_wait_asynccnt)
    __builtin_amdgcn_s_wait_asynccnt(0);
#else
    asm volatile("s_wait_asynccnt 0" ::: "memory");
#endif
    __syncthreads();

    // ---- Compute: one whole 2x2 patch per thread-iteration.
    //      idx = img_local*196 + patch  =>  output offset = grp_base + idx*4,
    //      so float4 stores are perfectly dense/coalesced. ----
    for (int idx = tid; idx < GROUP_PATCHES; idx += THREADS) {
        long long oofs = grp_base + (long long)idx * 4;
        if (oofs + 4 > total_floats) break;

        int img = idx / PATCHES;          // 0..7
        int t   = idx - img * PATCHES;    // 0..195
        int i   = t / 14;
        int j   = t - i * 14;

        const float* base = &smem[img * IMG_FLOATS];
        int r0 = (2 * i) * 28 + 2 * j;    // even float offset -> 8B aligned
        // Two 8B LDS reads, 112 B apart -> compiler emits one DS_LOAD_2ADDR_B64.
        float2 p0 = *(const float2*)(base + r0);
        float2 p1 = *(const float2*)(base + r0 + 28);

        // z = C_q * cos(p) + S_q * sin(p); pixels in [0,1] => hardware
        // v_cos_f32 / v_sin_f32 are accurate, no range reduction needed.
        float z00 = fmaf(C0, __cosf(p0.x), S0 * __sinf(p0.x));
        float z01 = fmaf(C1, __cosf(p0.y), S1 * __sinf(p0.y));
        float z10 = fmaf(C2, __cosf(p1.x), S2 * __sinf(p1.x));
        float z11 = fmaf(C3, __cosf(p1.y), S3 * __sinf(p1.y));

        v4f o = { z00, z00 * z01, z10, z10 * z11 };
        // Streamed once, > L2 capacity -> non-temporal B128 store.
        __builtin_nontemporal_store(o, (v4f*)(out + oofs));
    }
}

extern "C" void kernel_launch(void* const* d_in, const int* in_sizes, int n_in,
                              void* d_out, int out_size, void* d_ws, size_t ws_size,
                              hipStream_t stream) {
    const float* x      = (const float*)d_in[0];   // (B,1,28,28) f32
    const float* params = (const float*)d_in[1];   // (2,4) f32
    float*       out    = (float*)d_out;           // (B,784) f32
    float*       cs     = (float*)d_ws;            // 8 floats of scratch

    const int n_img = in_sizes[0] / IMG_FLOATS;                 // 65536
    const long long total_floats = (long long)n_img * IMG_FLOATS;

    quanv_precompute<<<1, 32, 0, stream>>>(params, cs);

    const int blocks = (n_img + IMG_PER_BLOCK - 1) / IMG_PER_BLOCK;  // 8192
    quanv_main<<<blocks, THREADS, 0, stream>>>(x, cs, out, total_floats);
}